// exp_LL_custom_V_2181843386538
// MI455X (gfx1250) — compile-verified
//
#include <hip/hip_runtime.h>

#define WPB 8          // waves per block (wave32)
#define SQUARINGS 8
#define DT_STEP 0.01f
#define INV_POW2 (1.0f / 256.0f)   // 2^-SQUARINGS

typedef __attribute__((ext_vector_type(2))) float v2f;
typedef __attribute__((ext_vector_type(8))) float v8f;

// ---------------------------------------------------------------------------
// Pauli helpers (complex elements of sigma_0..sigma_3 and two-qubit strings)
// ---------------------------------------------------------------------------
__device__ __forceinline__ void pauli_elem(int s, int r, int c, float& re, float& im) {
  re = 0.f; im = 0.f;
  switch (s) {
    case 0: re = (r == c) ? 1.f : 0.f; break;                                   // I
    case 1: re = (r != c) ? 1.f : 0.f; break;                                   // X
    case 2: im = (r == 0 && c == 1) ? -1.f : ((r == 1 && c == 0) ? 1.f : 0.f);  // Y
            break;
    default: re = (r == c) ? (r == 0 ? 1.f : -1.f) : 0.f; break;                // Z
  }
}

// P_code = sigma_{code>>2} (x) sigma_{code&3}, code in 1..15 ; 4x4 element (row,col)
__device__ __forceinline__ void pstring_elem(int code, int row, int col, float& re, float& im) {
  float ar, ai, br, bi;
  pauli_elem(code >> 2, row >> 1, col >> 1, ar, ai);
  pauli_elem(code & 3,  row & 1,  col & 1,  br, bi);
  re = ar * br - ai * bi;
  im = ar * bi + ai * br;
}

// ---------------------------------------------------------------------------
// Setup: build the three pre-scaled 16x16 generator matrices in d_ws:
//   cmats[0..255]   = (DT/2^s) * H           (from omega)
//   cmats[256..511] = (DT/2^s) * 100 * H_int (from omega_int)
//   cmats[512..767] = (DT/2^s) * dissipative (from v_x, v_y)
// One workgroup; f,d (15^3 each) and c_re/c_im live in LDS.
// ---------------------------------------------------------------------------
__global__ __launch_bounds__(256) void setup_liouvillian(
    const float* __restrict__ vx, const float* __restrict__ vy,
    const float* __restrict__ om, const float* __restrict__ omi,
    float* __restrict__ cmats)
{
  __shared__ float F[3375];   // f_abc
  __shared__ float Dd[3375];  // d_abc
  __shared__ float CRE[225];
  __shared__ float CIM[225];
  const int t = threadIdx.x;

  // Stage A: structure constants  Tr(Pa Pb Pc) = 4 (d + i f)
  for (int e = t; e < 3375; e += 256) {
    const int a = e / 225;
    const int rem = e - a * 225;
    const int b = rem / 15;
    const int c = rem - b * 15;
    const int pa = a + 1, pb = b + 1, pc = c + 1;
    float tr_re = 0.f, tr_im = 0.f;
    for (int i = 0; i < 4; ++i)
      for (int j = 0; j < 4; ++j) {
        float ar, ai; pstring_elem(pa, i, j, ar, ai);
        if (ar == 0.f && ai == 0.f) continue;
        for (int k = 0; k < 4; ++k) {
          float br, bi; pstring_elem(pb, j, k, br, bi);
          float cr, ci; pstring_elem(pc, k, i, cr, ci);
          const float abr = ar * br - ai * bi;
          const float abi = ar * bi + ai * br;
          tr_re += abr * cr - abi * ci;
          tr_im += abr * ci + abi * cr;
        }
      }
    F[e]  = 0.25f * tr_im;
    Dd[e] = 0.25f * tr_re;
  }

  // Stage B: Kossakowski matrix c = v^H v  (re / im parts)
  if (t < 225) {
    const int i = t / 15, j = t - i * 15;
    float re = 0.f, im = 0.f;
    for (int k = 0; k < 15; ++k) {
      const float xki = vx[k * 15 + i], xkj = vx[k * 15 + j];
      const float yki = vy[k * 15 + i], ykj = vy[k * 15 + j];
      re += xki * xkj + yki * ykj;
      im += xki * ykj - yki * xkj;
    }
    CRE[t] = re; CIM[t] = im;
  }
  __syncthreads();

  // Stage C: assemble 16x16 matrices; thread t = row*16 + col
  const int r = t >> 4, c = t & 15;
  const float scale = DT_STEP * INV_POW2;
  float hval = 0.f, hintval = 0.f, dval = 0.f;

  if (r >= 1 && c >= 1) {
    const int jj = r - 1;   // H16[1+j][1+i] = 4 * sum_k f[i][j][k] * w[k]
    const int ii = c - 1;
    float s1 = 0.f, s2 = 0.f;
    for (int k = 0; k < 15; ++k) {
      const float fv = F[ii * 225 + jj * 15 + k];
      s1 += fv * om[k];
      s2 += fv * omi[k];
    }
    hval = 4.f * s1;
    hintval = 4.f * s2;

    // d_super[m][n]
    const int m = r - 1, nn = c - 1;
    float acc = 0.f;
    for (int i = 0; i < 15; ++i)
      for (int j = 0; j < 15; ++j) {
        const float cre = CRE[i * 15 + j];
        const float cim = CIM[i * 15 + j];
        float s = 0.f;
        for (int k = 0; k < 15; ++k) {
          const float fmjk = F[m  * 225 + j * 15 + k];
          const float fmik = F[m  * 225 + i * 15 + k];
          const float fnik = F[nn * 225 + i * 15 + k];
          const float fnjk = F[nn * 225 + j * 15 + k];
          const float dnik = Dd[nn * 225 + i * 15 + k];
          const float dnjk = Dd[nn * 225 + j * 15 + k];
          s += cre * (fmjk * fnik + fmik * fnjk)
             + cim * (fmjk * dnik - fmik * dnjk);
        }
        acc += s;
      }
    dval = -4.f * acc;
  } else if (c == 0 && r >= 1) {
    // tr_id[m] = -4 * sum_{i,j} f[i][m][j] * c_im[i][j]
    const int m = r - 1;
    float acc = 0.f;
    for (int i = 0; i < 15; ++i)
      for (int j = 0; j < 15; ++j)
        acc += F[i * 225 + m * 15 + j] * CIM[i * 15 + j];
    dval = -4.f * acc;
  }

  cmats[t]       = scale * hval;
  cmats[256 + t] = scale * 100.f * hintval;
  cmats[512 + t] = scale * dval;
}

// ---------------------------------------------------------------------------
// Main: one wave per sample. expm via order-4 Taylor + 8 squarings, all
// 16x16x16 matmuls done as 4 chained V_WMMA_F32_16X16X4_F32.
// C/D layout: lane = col + 16*(row>=8), vgpr r = row%8.
// A-slices read from a column-major LDS copy, B-slices from a row-major copy;
// both use address (4q + r + 2h)*16 + (lane%16)  (conflict-free).
// ---------------------------------------------------------------------------
__global__ __launch_bounds__(256) void expm_apply(
    const float* __restrict__ x, const float* __restrict__ pot,
    const float* __restrict__ cmats, float* __restrict__ out, int Bn)
{
  __shared__ float cm[768];
  __shared__ float lds[WPB * 512];

  const int tid = threadIdx.x;
  for (int i = tid; i < 768; i += 256) cm[i] = cmats[i];
  __syncthreads();

  const int lane = tid & 31;
  const int wave = tid >> 5;
  const int b = blockIdx.x * WPB + wave;
  if (b >= Bn) return;                 // wave-uniform: EXEC stays all-ones

  const int h = lane >> 4;             // row-half
  const int n = lane & 15;             // column index in C/D layout
  float* ldsR = &lds[wave * 512];      // row-major tile
  float* ldsC = ldsR + 256;            // column-major tile

  const float p  = pot[b];
  const float p2 = p * p;

  // Build Ms = (DT/2^s) * (H + 100 p H_int + p^2 D) in C/D layout
  float ms[8], iden[8];
#pragma unroll
  for (int r = 0; r < 8; ++r) {
    const int row = r + 8 * h;
    const int idx = row * 16 + n;
    ms[r]   = cm[idx] + p * cm[256 + idx] + p2 * cm[512 + idx];
    iden[r] = (row == n) ? 1.f : 0.f;
  }

  // A-layout slices of Ms via column-major LDS round-trip (reused in Taylor)
#pragma unroll
  for (int r = 0; r < 8; ++r) ldsC[n * 16 + 8 * h + r] = ms[r];
  asm volatile("" ::: "memory");
  float aS[8];
#pragma unroll
  for (int q = 0; q < 4; ++q) {
    const int k0 = (4 * q + 2 * h) * 16 + n;
    aS[q * 2 + 0] = ldsC[k0];
    aS[q * 2 + 1] = ldsC[k0 + 16];
  }
  asm volatile("" ::: "memory");

  // Horner Taylor: T = I + A(I + A/2(I + A/3(I + A/4)))
  float tmat[8];
#pragma unroll
  for (int r = 0; r < 8; ++r) tmat[r] = iden[r] + 0.25f * ms[r];

  const float coefs[3] = {1.f / 3.f, 0.5f, 1.f};
#pragma unroll
  for (int stp = 0; stp < 3; ++stp) {
    const float cf = coefs[stp];
#pragma unroll
    for (int r = 0; r < 8; ++r) ldsR[(r + 8 * h) * 16 + n] = tmat[r];
    asm volatile("" ::: "memory");
    v8f acc;
#pragma unroll
    for (int r = 0; r < 8; ++r) acc[r] = iden[r];
#pragma unroll
    for (int q = 0; q < 4; ++q) {
      const int k0 = (4 * q + 2 * h) * 16 + n;
      v2f av, bv;
      av[0] = cf * aS[q * 2 + 0];
      av[1] = cf * aS[q * 2 + 1];
      bv[0] = ldsR[k0];
      bv[1] = ldsR[k0 + 16];
      acc = __builtin_amdgcn_wmma_f32_16x16x4_f32(false, av, false, bv,
                                                  (short)0, acc, false, false);
    }
#pragma unroll
    for (int r = 0; r < 8; ++r) tmat[r] = acc[r];
    asm volatile("" ::: "memory");
  }

  // Repeated squaring: E <- E*E, 8 times
  for (int sq = 0; sq < SQUARINGS; ++sq) {
#pragma unroll
    for (int r = 0; r < 8; ++r) {
      ldsR[(r + 8 * h) * 16 + n] = tmat[r];
      ldsC[n * 16 + 8 * h + r]   = tmat[r];
    }
    asm volatile("" ::: "memory");
    v8f acc;
#pragma unroll
    for (int r = 0; r < 8; ++r) acc[r] = 0.f;
#pragma unroll
    for (int q = 0; q < 4; ++q) {
      const int k0 = (4 * q + 2 * h) * 16 + n;
      v2f av, bv;
      av[0] = ldsC[k0];
      av[1] = ldsC[k0 + 16];
      bv[0] = ldsR[k0];
      bv[1] = ldsR[k0 + 16];
      acc = __builtin_amdgcn_wmma_f32_16x16x4_f32(false, av, false, bv,
                                                  (short)0, acc, false, false);
    }
#pragma unroll
    for (int r = 0; r < 8; ++r) tmat[r] = acc[r];
    asm volatile("" ::: "memory");
  }

  // y = E @ [1, x_b] ; reduce across the 16 lanes of each half
  const float xv = (n == 0) ? 1.f : x[(size_t)b * 15 + (n - 1)];
  float yv[8];
#pragma unroll
  for (int r = 0; r < 8; ++r) yv[r] = tmat[r] * xv;
#pragma unroll
  for (int m = 1; m <= 8; m <<= 1) {
#pragma unroll
    for (int r = 0; r < 8; ++r) yv[r] += __shfl_xor(yv[r], m, 32);
  }

  // out = y[1:]: rows 1..7 live in half 0, rows 8..15 in half 1
  if (lane == 0) {
#pragma unroll
    for (int r = 1; r < 8; ++r) out[(size_t)b * 15 + (r - 1)] = yv[r];
  } else if (lane == 16) {
#pragma unroll
    for (int r = 0; r < 8; ++r) out[(size_t)b * 15 + 7 + r] = yv[r];
  }
}

// ---------------------------------------------------------------------------
extern "C" void kernel_launch(void* const* d_in, const int* in_sizes, int n_in,
                              void* d_out, int out_size, void* d_ws, size_t ws_size,
                              hipStream_t stream) {
  const float* x   = (const float*)d_in[0];
  const float* pot = (const float*)d_in[1];
  const float* vx  = (const float*)d_in[2];
  const float* vy  = (const float*)d_in[3];
  const float* om  = (const float*)d_in[4];
  const float* omi = (const float*)d_in[5];
  float* out   = (float*)d_out;
  float* cmats = (float*)d_ws;          // 768 floats
  const int Bn = in_sizes[1];           // potential: [B]

  setup_liouvillian<<<1, 256, 0, stream>>>(vx, vy, om, omi, cmats);

  const int blocks = (Bn + WPB - 1) / WPB;
  expm_apply<<<blocks, 256, 0, stream>>>(x, pot, cmats, out, Bn);
}